// Multihead_Attention_67800353734967
// MI455X (gfx1250) — compile-verified
//
#include <hip/hip_runtime.h>

// ---------------------------------------------------------------------------
// Multi-head "attention" (no softmax) for MI455X / gfx1250, fp32 WMMA path.
//
//   Q = x@Wq^T+bq ; K = x@Wk^T+bk ; V = x@Wv^T+bv      (4096x1024 each)
//   per (h,b) slab (raw row-major head split, 32 slabs of [2048,64]):
//       KtV = K^T V * NORM                (64x64)       <- associativity: no softmax
//       AO  = Q @ KtV                     (2048x64)
//   out = AO@Wo^T + bo                                  (4096x1024)
//
// All GEMMs run on V_WMMA_F32_16X16X4_F32 (fp32 in, fp32 acc).
// Big-GEMM wave tile is 64x64: 16 WMMAs per 8 fragment loads (2:1 ratio,
// 16 FLOP/byte register reuse) — the whole working set lives in the 192MB L2,
// so the limiter is issue/cache bandwidth, not HBM.
// ---------------------------------------------------------------------------

typedef float v2f __attribute__((ext_vector_type(2)));
typedef float v8f __attribute__((ext_vector_type(8)));

#define D_MODEL 1024
#define B_SZ    2
#define S_LEN   2048
#define M_TOT   (B_SZ * S_LEN)     // 4096 rows
#define DH      64                 // head dim
#define N_HB    32                 // H * B slabs
#define NORMF   0.03125f           // 1/sqrt(1024)

__device__ __forceinline__ v8f wmma4(v2f a, v2f b, v8f c) {
  // 8-arg form: (neg_a, A, neg_b, B, c_mod, C, reuse_a, reuse_b)
  return __builtin_amdgcn_wmma_f32_16x16x4_f32(
      false, a, false, b, (short)0, c, false, false);
}

// ---------------------------------------------------------------------------
// C[m][n] = sum_k A[m][k]*W[n][k] + bias[n]
// A: [M, 1024] row-major, W: [1024, 1024] row-major ([n][k]), C: [M, 1024].
// Block = 256 thr (8 waves) covering 128M x 256N; wave tile 64x64
// (4 A-frags x 4 B-frags -> 16 WMMAs per 8 float2 loads per K-step of 4).
// A-frag (16x4 f32): lane l -> m = l&15, k-pair = 2*(l>>4)      (float2 load)
// B-frag (4x16 f32): lane l -> n = l&15, k-pair = 2*(l>>4); B[k][n] = W[n][k]
//   => identical addressing pattern on W.
// ---------------------------------------------------------------------------
__global__ __launch_bounds__(256) void gemm_bias_kernel(
    const float* __restrict__ A, const float* __restrict__ W,
    const float* __restrict__ bias, float* __restrict__ C) {
  const int lane = threadIdx.x & 31;
  const int wave = threadIdx.x >> 5;
  const int wm = wave >> 2;                 // 0..1
  const int wn = wave & 3;                  // 0..3
  const int m0 = blockIdx.x * 128 + wm * 64;
  const int n0 = blockIdx.y * 256 + wn * 64;

  const int lr = lane & 15;
  const int lk = (lane >> 4) * 2;

  const float* ap[4];
  const float* bp[4];
#pragma unroll
  for (int i = 0; i < 4; ++i) {
    ap[i] = A + (size_t)(m0 + 16 * i + lr) * D_MODEL + lk;
    bp[i] = W + (size_t)(n0 + 16 * i + lr) * D_MODEL + lk;
  }

  v8f acc[4][4];
#pragma unroll
  for (int i = 0; i < 4; ++i)
#pragma unroll
    for (int j = 0; j < 4; ++j) acc[i][j] = v8f{};

#pragma unroll 2
  for (int k = 0; k < D_MODEL; k += 4) {
    v2f a[4], b[4];
#pragma unroll
    for (int i = 0; i < 4; ++i) a[i] = *(const v2f*)(ap[i] + k);
#pragma unroll
    for (int j = 0; j < 4; ++j) b[j] = *(const v2f*)(bp[j] + k);
#pragma unroll
    for (int i = 0; i < 4; ++i)
#pragma unroll
      for (int j = 0; j < 4; ++j)
        acc[i][j] = wmma4(a[i], b[j], acc[i][j]);
  }

  // D layout: vgpr r holds M = 8*(lane>>4) + r, N = lane&15 within the tile.
  const int rb = (lane >> 4) * 8;
  float bv[4];
#pragma unroll
  for (int j = 0; j < 4; ++j) bv[j] = bias[n0 + 16 * j + lr];

#pragma unroll
  for (int i = 0; i < 4; ++i) {
#pragma unroll
    for (int r = 0; r < 8; ++r) {
      const size_t row = (size_t)(m0 + 16 * i + rb + r) * D_MODEL;
#pragma unroll
      for (int j = 0; j < 4; ++j)
        C[row + n0 + 16 * j + lr] = acc[i][j][r] + bv[j];
    }
  }
}

// ---------------------------------------------------------------------------
// Per (h,b) slab: KtV[d1][d2] = NORM * sum_s K[s][d1] * V[s][d2]   (64x64)
// One block per slab; wave w handles n-tile (w&3) and m-tiles 2*(w>>2)+{0,1}
// sharing one B fragment per K-step.
// ---------------------------------------------------------------------------
__global__ __launch_bounds__(256) void ktv_kernel(
    const float* __restrict__ Kbuf, const float* __restrict__ Vbuf,
    float* __restrict__ KtV) {
  const int hb = blockIdx.x;
  const float* Kh = Kbuf + (size_t)hb * S_LEN * DH;
  const float* Vh = Vbuf + (size_t)hb * S_LEN * DH;
  float* Ch = KtV + (size_t)hb * DH * DH;

  const int lane = threadIdx.x & 31;
  const int wave = threadIdx.x >> 5;
  const int n0 = (wave & 3) * 16;
  const int m0 = (wave >> 2) * 32;
  const int lr = lane & 15;
  const int lk = (lane >> 4) * 2;

  v8f acc0 = {}, acc1 = {};
#pragma unroll 4
  for (int k = 0; k < S_LEN; k += 4) {
    const int kk = k + lk;
    v2f a0, a1, b;
    // A = K^T: A[m][s] = K[s][m]  (stride-DH gathers)
    a0.x = Kh[(size_t)kk * DH + m0 + lr];
    a0.y = Kh[(size_t)(kk + 1) * DH + m0 + lr];
    a1.x = Kh[(size_t)kk * DH + m0 + 16 + lr];
    a1.y = Kh[(size_t)(kk + 1) * DH + m0 + 16 + lr];
    b.x  = Vh[(size_t)kk * DH + n0 + lr];
    b.y  = Vh[(size_t)(kk + 1) * DH + n0 + lr];
    acc0 = wmma4(a0, b, acc0);
    acc1 = wmma4(a1, b, acc1);
  }

  const int rb = (lane >> 4) * 8;
#pragma unroll
  for (int r = 0; r < 8; ++r) {
    Ch[(size_t)(m0 + rb + r) * DH + n0 + lr]      = acc0[r] * NORMF;
    Ch[(size_t)(m0 + 16 + rb + r) * DH + n0 + lr] = acc1[r] * NORMF;
  }
}

// ---------------------------------------------------------------------------
// Per (h,b) slab: AO[q][d] = sum_k Q[q][k] * KtV[k][d]   (2048x64, K=64)
// Wave tile 16x64 (4 n-tiles); block covers 128 q-rows.
// ---------------------------------------------------------------------------
__global__ __launch_bounds__(256) void attn_kernel(
    const float* __restrict__ Qbuf, const float* __restrict__ KtV,
    float* __restrict__ Obuf) {
  const int hb = blockIdx.y;
  const float* Qh = Qbuf + (size_t)hb * S_LEN * DH;
  const float* Ch = KtV + (size_t)hb * DH * DH;
  float* Oh = Obuf + (size_t)hb * S_LEN * DH;

  const int lane = threadIdx.x & 31;
  const int wave = threadIdx.x >> 5;
  const int q0 = blockIdx.x * 128 + wave * 16;
  const int lr = lane & 15;
  const int lk = (lane >> 4) * 2;

  v8f acc[4] = {v8f{}, v8f{}, v8f{}, v8f{}};
#pragma unroll
  for (int k = 0; k < DH; k += 4) {
    const int kk = k + lk;
    v2f a = *(const v2f*)(Qh + (size_t)(q0 + lr) * DH + kk);
#pragma unroll
    for (int j = 0; j < 4; ++j) {
      v2f b;
      b.x = Ch[(size_t)kk * DH + j * 16 + lr];
      b.y = Ch[(size_t)(kk + 1) * DH + j * 16 + lr];
      acc[j] = wmma4(a, b, acc[j]);
    }
  }

  const int rb = (lane >> 4) * 8;
#pragma unroll
  for (int j = 0; j < 4; ++j)
#pragma unroll
    for (int r = 0; r < 8; ++r)
      Oh[(size_t)(q0 + rb + r) * DH + j * 16 + lr] = acc[j][r];
}

// ---------------------------------------------------------------------------
extern "C" void kernel_launch(void* const* d_in, const int* in_sizes, int n_in,
                              void* d_out, int out_size, void* d_ws, size_t ws_size,
                              hipStream_t stream) {
  (void)in_sizes; (void)n_in; (void)out_size; (void)ws_size;
  const float* x  = (const float*)d_in[0];
  // d_in[1] (y) is only used for its shape in the reference.
  const float* Wq = (const float*)d_in[2];
  const float* bq = (const float*)d_in[3];
  const float* Wk = (const float*)d_in[4];
  const float* bk = (const float*)d_in[5];
  const float* Wv = (const float*)d_in[6];
  const float* bv = (const float*)d_in[7];
  const float* Wo = (const float*)d_in[8];
  const float* bo = (const float*)d_in[9];
  float* out = (float*)d_out;

  char* ws = (char*)d_ws;
  const size_t MB = 1024ull * 1024ull;
  float* Q   = (float*)(ws);              // 16 MB
  float* Kb  = (float*)(ws + 16 * MB);    // 16 MB
  float* Vb  = (float*)(ws + 32 * MB);    // 16 MB
  float* AO  = (float*)(ws + 48 * MB);    // 16 MB
  float* KtV = (float*)(ws + 64 * MB);    // 512 KB

  dim3 gproj(M_TOT / 128, D_MODEL / 256);  // 32 x 4 blocks, 8 waves each
  gemm_bias_kernel<<<gproj, 256, 0, stream>>>(x, Wq, bq, Q);
  gemm_bias_kernel<<<gproj, 256, 0, stream>>>(x, Wk, bk, Kb);
  gemm_bias_kernel<<<gproj, 256, 0, stream>>>(x, Wv, bv, Vb);

  ktv_kernel<<<N_HB, 256, 0, stream>>>(Kb, Vb, KtV);
  attn_kernel<<<dim3(S_LEN / 128, N_HB), 256, 0, stream>>>(Q, KtV, AO);

  gemm_bias_kernel<<<gproj, 256, 0, stream>>>(AO, Wo, bo, out);
}